// GAT3_79671643340946
// MI455X (gfx1250) — compile-verified
//
#include <hip/hip_runtime.h>
#include <hip/hip_bf16.h>
#include <math.h>

#define F_IN 64
#define HEADS 4
#define HID 16
#define CH1 64   // HEADS*HID
#define NEG_SLOPE 0.2f

typedef float v2f __attribute__((ext_vector_type(2)));
typedef float v8f __attribute__((ext_vector_type(8)));

// ---- helpers -------------------------------------------------------------

// monotone float <-> uint32 key, so unsigned atomicMax == float max
__device__ __forceinline__ unsigned fkey(float f) {
  unsigned u = __float_as_uint(f);
  return (u & 0x80000000u) ? ~u : (u | 0x80000000u);
}
__device__ __forceinline__ float unfkey(unsigned k) {
  unsigned u = (k & 0x80000000u) ? (k & 0x7fffffffu) : ~k;
  return __uint_as_float(u);
}
__device__ __forceinline__ float lrelu(float v) {
  return v > 0.f ? v : NEG_SLOPE * v;
}
// edge e in [0,E) is a real edge; e in [E,E+N) is self-loop (e-E, e-E)
__device__ __forceinline__ void edge_nodes(const int* __restrict__ ei, int e,
                                           int E, int& s, int& d) {
  if (e < E) { s = ei[e]; d = ei[E + e]; } else { s = e - E; d = e - E; }
}

// ---- 0) transpose W1 -> WT[col][k] so B fragments are contiguous b64 -----

__global__ void gat_w1t(const float* __restrict__ W, float* __restrict__ WT) {
  int i = blockIdx.x * blockDim.x + threadIdx.x;   // i = k*64 + c
  if (i < F_IN * CH1) {
    int k = i >> 6, c = i & 63;
    WT[c * F_IN + k] = W[i];
  }
}

// ---- 1) h1 = X @ W1 via fp32 WMMA, fused attention logits ---------------
// One wave computes a 16-node x 64-channel tile; head h == column tile nt.

__global__ __launch_bounds__(32) void gat_gemm1_wmma(
    const float* __restrict__ X, const float* __restrict__ WT,
    const float* __restrict__ attS, const float* __restrict__ attD,
    float* __restrict__ H, float* __restrict__ as1, float* __restrict__ ad1,
    int N) {
  const int lane = threadIdx.x;
  const int ln16 = lane & 15;
  const int hi   = lane >> 4;               // 0 or 1 (lane half)
  const int tileBase = blockIdx.x * 16;
  const bool fullTile = (tileBase + 16 <= N);

  int row = tileBase + ln16;
  if (row >= N) row = N - 1;                // clamp: keep EXEC all-ones for WMMA
  const float* xrow = X + (long long)row * F_IN;

  const int kb = hi * 2;                    // A/B frag: lanes 16-31 hold K+2,K+3
  v2f afrag[16];
#pragma unroll
  for (int kk = 0; kk < 16; ++kk)
    afrag[kk] = *(const v2f*)(xrow + kk * 4 + kb);

  const int mb = hi * 8;                    // D layout: lanes 16-31 hold M+8
#pragma unroll
  for (int nt = 0; nt < 4; ++nt) {
    const int col = nt * 16 + ln16;
    const float* wtcol = WT + col * F_IN;   // transposed: contiguous in K
    v8f acc = {0.f, 0.f, 0.f, 0.f, 0.f, 0.f, 0.f, 0.f};
#pragma unroll
    for (int kk = 0; kk < 16; ++kk) {
      v2f b = *(const v2f*)(wtcol + kk * 4 + kb);   // single b64 load
      acc = __builtin_amdgcn_wmma_f32_16x16x4_f32(false, afrag[kk], false, b,
                                                  (short)0, acc, false, false);
    }
    // store h1 tile (uniform scalar branch; hot path unguarded)
    if (fullTile) {
#pragma unroll
      for (int r = 0; r < 8; ++r)
        H[(long long)(tileBase + mb + r) * CH1 + col] = acc[r];
    } else {
#pragma unroll
      for (int r = 0; r < 8; ++r) {
        int orow = tileBase + mb + r;
        if (orow < N) H[(long long)orow * CH1 + col] = acc[r];
      }
    }
    // fused attention logits for head nt: 16-lane xor-shuffle reduction
    const float aSv = attS[nt * HID + ln16];
    const float aDv = attD[nt * HID + ln16];
#pragma unroll
    for (int r = 0; r < 8; ++r) {
      float sv = acc[r] * aSv;
      float dv = acc[r] * aDv;
      sv += __shfl_xor(sv, 8); dv += __shfl_xor(dv, 8);
      sv += __shfl_xor(sv, 4); dv += __shfl_xor(dv, 4);
      sv += __shfl_xor(sv, 2); dv += __shfl_xor(dv, 2);
      sv += __shfl_xor(sv, 1); dv += __shfl_xor(dv, 1);
      const int orow = tileBase + mb + r;   // lanes 0 & 16 own the two halves
      if (ln16 == 0 && orow < N) {
        as1[orow * HEADS + nt] = sv;
        ad1[orow * HEADS + nt] = dv;
      }
    }
  }
}

// ---- 2) init accumulators (workspace is poisoned, must zero) -------------

__global__ void gat_init1(float* __restrict__ out1, float* __restrict__ den1,
                          unsigned* __restrict__ m1k, int N) {
  long long i = (long long)blockIdx.x * blockDim.x + threadIdx.x;
  if (i < (long long)N * CH1) out1[i] = 0.f;
  if (i < (long long)N * HEADS) { den1[i] = 0.f; m1k[i] = 0u; }
}

// ---- 3-5) layer-1 edge passes --------------------------------------------

__global__ void gat_edge_max1(const int* __restrict__ ei,
                              const float* __restrict__ as1,
                              const float* __restrict__ ad1,
                              float* __restrict__ e1,
                              unsigned* __restrict__ m1k, int E, int N) {
  long long gid = (long long)blockIdx.x * blockDim.x + threadIdx.x;
  if (gid >= (long long)(E + N) * HEADS) return;
  int e = (int)(gid >> 2), h = (int)(gid & 3);
  int s, d; edge_nodes(ei, e, E, s, d);
  float v = lrelu(as1[s * HEADS + h] + ad1[d * HEADS + h]);
  e1[gid] = v;
  atomicMax(&m1k[d * HEADS + h], fkey(v));
}

__global__ void gat_edge_sum1(const int* __restrict__ ei,
                              float* __restrict__ e1,
                              const unsigned* __restrict__ m1k,
                              float* __restrict__ den1, int E, int N) {
  long long gid = (long long)blockIdx.x * blockDim.x + threadIdx.x;
  if (gid >= (long long)(E + N) * HEADS) return;
  int e = (int)(gid >> 2), h = (int)(gid & 3);
  int s, d; edge_nodes(ei, e, E, s, d);
  float ex = __expf(e1[gid] - unfkey(m1k[d * HEADS + h]));
  e1[gid] = ex;
  atomicAdd(&den1[d * HEADS + h], ex);
}

__global__ void gat_edge_agg1(const int* __restrict__ ei,
                              const float* __restrict__ e1,
                              const float* __restrict__ den1,
                              const float* __restrict__ h1,
                              float* __restrict__ out1, int E, int N) {
  long long gid = (long long)blockIdx.x * blockDim.x + threadIdx.x;
  if (gid >= (long long)(E + N) * CH1) return;
  int e = (int)(gid >> 6), c = (int)(gid & 63), h = c >> 4;
  int s, d; edge_nodes(ei, e, E, s, d);
  float alpha = e1[(long long)e * HEADS + h] / (den1[d * HEADS + h] + 1e-16f);
  atomicAdd(&out1[(long long)d * CH1 + c], h1[(long long)s * CH1 + c] * alpha);
}

// ---- 6) ReLU+bias, layer-2 transform, logits, output init ----------------

__global__ void gat_node2(const float* __restrict__ out1,
                          const float* __restrict__ b1,
                          const float* __restrict__ W2,
                          const float* __restrict__ aS2,
                          const float* __restrict__ aD2,
                          const float* __restrict__ b2,
                          float* __restrict__ h2, float* __restrict__ as2,
                          float* __restrict__ ad2, unsigned* __restrict__ m2k,
                          float* __restrict__ den2, float* __restrict__ out,
                          int N) {
  int n = blockIdx.x * blockDim.x + threadIdx.x;
  if (n >= N) return;
  const float* xr = out1 + (long long)n * CH1;
  float a0 = 0.f, a1 = 0.f;
#pragma unroll 8
  for (int c = 0; c < CH1; ++c) {
    float v = xr[c] + b1[c];
    v = v > 0.f ? v : 0.f;                 // ReLU between layers
    a0 += v * W2[c * 2 + 0];
    a1 += v * W2[c * 2 + 1];
  }
  h2[n * 2 + 0] = a0;
  h2[n * 2 + 1] = a1;
  as2[n] = a0 * aS2[0] + a1 * aS2[1];
  ad2[n] = a0 * aD2[0] + a1 * aD2[1];
  m2k[n] = 0u;
  den2[n] = 0.f;
  out[n * 2 + 0] = b2[0];                  // bias-seeded output accumulator
  out[n * 2 + 1] = b2[1];
}

// ---- 7-9) layer-2 edge passes (1 head, 2 channels) -----------------------

__global__ void gat_edge_max2(const int* __restrict__ ei,
                              const float* __restrict__ as2,
                              const float* __restrict__ ad2,
                              float* __restrict__ e2,
                              unsigned* __restrict__ m2k, int E, int N) {
  int e = blockIdx.x * blockDim.x + threadIdx.x;
  if (e >= E + N) return;
  int s, d; edge_nodes(ei, e, E, s, d);
  float v = lrelu(as2[s] + ad2[d]);
  e2[e] = v;
  atomicMax(&m2k[d], fkey(v));
}

__global__ void gat_edge_sum2(const int* __restrict__ ei,
                              float* __restrict__ e2,
                              const unsigned* __restrict__ m2k,
                              float* __restrict__ den2, int E, int N) {
  int e = blockIdx.x * blockDim.x + threadIdx.x;
  if (e >= E + N) return;
  int s, d; edge_nodes(ei, e, E, s, d);
  float ex = __expf(e2[e] - unfkey(m2k[d]));
  e2[e] = ex;
  atomicAdd(&den2[d], ex);
}

__global__ void gat_edge_agg2(const int* __restrict__ ei,
                              const float* __restrict__ e2,
                              const float* __restrict__ den2,
                              const float* __restrict__ h2,
                              float* __restrict__ out, int E, int N) {
  long long gid = (long long)blockIdx.x * blockDim.x + threadIdx.x;
  if (gid >= (long long)(E + N) * 2) return;
  int e = (int)(gid >> 1), c = (int)(gid & 1);
  int s, d; edge_nodes(ei, e, E, s, d);
  float alpha = e2[e] / (den2[d] + 1e-16f);
  atomicAdd(&out[(long long)d * 2 + c], h2[(long long)s * 2 + c] * alpha);
}

// ---- launch --------------------------------------------------------------

extern "C" void kernel_launch(void* const* d_in, const int* in_sizes, int n_in,
                              void* d_out, int out_size, void* d_ws,
                              size_t ws_size, hipStream_t stream) {
  const float* X   = (const float*)d_in[0];
  const int*   EI  = (const int*)d_in[1];
  /* d_in[2] = edge_attr: ignored by GATConv when edge_dim is None */
  const float* W1  = (const float*)d_in[3];
  const float* AS1 = (const float*)d_in[4];
  const float* AD1 = (const float*)d_in[5];
  const float* B1  = (const float*)d_in[6];
  const float* W2  = (const float*)d_in[7];
  const float* AS2 = (const float*)d_in[8];
  const float* AD2 = (const float*)d_in[9];
  const float* B2  = (const float*)d_in[10];
  float* OUT = (float*)d_out;
  (void)n_in; (void)out_size; (void)ws_size;

  const int N  = in_sizes[0] / F_IN;
  const int E  = in_sizes[1] / 2;
  const int ET = E + N;

  // workspace carve-up (all 4-byte elements; ~94 MB total, L2-resident)
  float* ws     = (float*)d_ws;
  float* h1     = ws;                 ws += (size_t)N * CH1;
  float* as1    = ws;                 ws += (size_t)N * HEADS;
  float* ad1    = ws;                 ws += (size_t)N * HEADS;
  unsigned* m1k = (unsigned*)ws;      ws += (size_t)N * HEADS;
  float* den1   = ws;                 ws += (size_t)N * HEADS;
  float* e1     = ws;                 ws += (size_t)ET * HEADS;
  float* out1   = ws;                 ws += (size_t)N * CH1;
  float* h2     = ws;                 ws += (size_t)N * 2;
  float* as2    = ws;                 ws += (size_t)N;
  float* ad2    = ws;                 ws += (size_t)N;
  unsigned* m2k = (unsigned*)ws;      ws += (size_t)N;
  float* den2   = ws;                 ws += (size_t)N;
  float* e2     = ws;                 ws += (size_t)ET;
  float* W1T    = ws;                 ws += (size_t)F_IN * CH1;

  const int TPB = 256;
  gat_w1t<<<(F_IN * CH1 + TPB - 1) / TPB, TPB, 0, stream>>>(W1, W1T);
  gat_gemm1_wmma<<<(unsigned)((N + 15) / 16), 32, 0, stream>>>(
      X, W1T, AS1, AD1, h1, as1, ad1, N);

  long long ti = (long long)N * CH1;
  gat_init1<<<(unsigned)((ti + TPB - 1) / TPB), TPB, 0, stream>>>(out1, den1,
                                                                  m1k, N);
  long long t4 = (long long)ET * HEADS;
  gat_edge_max1<<<(unsigned)((t4 + TPB - 1) / TPB), TPB, 0, stream>>>(
      EI, as1, ad1, e1, m1k, E, N);
  gat_edge_sum1<<<(unsigned)((t4 + TPB - 1) / TPB), TPB, 0, stream>>>(
      EI, e1, m1k, den1, E, N);
  long long t64 = (long long)ET * CH1;
  gat_edge_agg1<<<(unsigned)((t64 + TPB - 1) / TPB), TPB, 0, stream>>>(
      EI, e1, den1, h1, out1, E, N);

  gat_node2<<<(unsigned)((N + TPB - 1) / TPB), TPB, 0, stream>>>(
      out1, B1, W2, AS2, AD2, B2, h2, as2, ad2, m2k, den2, OUT, N);

  gat_edge_max2<<<(unsigned)((ET + TPB - 1) / TPB), TPB, 0, stream>>>(
      EI, as2, ad2, e2, m2k, E, N);
  gat_edge_sum2<<<(unsigned)((ET + TPB - 1) / TPB), TPB, 0, stream>>>(
      EI, e2, m2k, den2, E, N);
  long long t2 = (long long)ET * 2;
  gat_edge_agg2<<<(unsigned)((t2 + TPB - 1) / TPB), TPB, 0, stream>>>(
      EI, e2, den2, h2, OUT, E, N);
}